// SpatioTemporalGraphBuilder_51238959841298
// MI455X (gfx1250) — compile-verified
//
#include <hip/hip_runtime.h>
#include <hip/hip_bf16.h>
#include <stdint.h>

// ---------------- problem constants (from reference) ----------------
#define GS        32            // GRID_SIZE
#define BATCH     16
#define CH        3
#define FRAMES    32
#define IMH       224
#define IMW       224
#define GH        7             // 224/32
#define GW        7
#define NPF       49            // nodes per frame
#define NODES     (FRAMES * NPF)         // 1568
#define NPATCH    (BATCH * NODES)        // 25088
#define PATCH_N   (CH * GS * GS)         // 3072 elements per patch
#define SORT_N    4096                   // padded (next pow2) for bitonic
#define NTHREADS  256                    // 8 wave32 per workgroup
#define CH_STRIDE (FRAMES * IMH * IMW)   // 1605632 elements between channels
#define E_TOTAL   15962                  // 32*312 spatial + 49*122 temporal

typedef unsigned int v4u __attribute__((ext_vector_type(4)));
typedef int          v8i __attribute__((ext_vector_type(8)));
typedef int          v4i __attribute__((ext_vector_type(4)));
typedef float        v2f __attribute__((ext_vector_type(2)));
typedef float        v8f __attribute__((ext_vector_type(8)));

// ---------------- TDM: one 3-D tile load per patch ----------------
// D# per CDNA5 ISA:
//  tile:  dim0=32 (contig floats), dim1=32 rows stride 224, dim2=3 ch stride 1605632
//  group0: count=1 | lds_addr | global_addr(57b) | type=2
// amdgpu-toolchain (clang-23) 6-arg builtin: (v4u g0, v8i g1, v4i g2, v4i g3, v8i extra, i32 cpol)
__device__ __forceinline__ void tdm_load_patch(const float* gsrc, uint32_t lds_byte_off) {
  uint64_t ga = (uint64_t)(uintptr_t)gsrc;
  v4u g0; v8i g1; v4i g2; v4i g3; v8i gx;
  g0[0] = 1u;                                             // count=1 (valid), user mode
  g0[1] = lds_byte_off;                                   // lds_addr [63:32]
  g0[2] = (uint32_t)ga;                                   // global_addr [95:64]
  g0[3] = (uint32_t)((ga >> 32) & 0x01FFFFFFu) | (2u << 30); // addr[56:32] | type=2
  g1[0] = (int)(2u << 16);                                // data_size=2 (4B), mask=0, no pad
  g1[1] = (int)((unsigned)GS << 16);                      // tensor_dim0 lo16 @bits[63:48]
  g1[2] = (int)((unsigned)GS << 16);                      // tensor_dim1 lo16 @bits[95:80]
  g1[3] = (int)((unsigned)GS << 16);                      // tile_dim0 @bits[127:112]
  g1[4] = (int)((unsigned)GS | ((unsigned)CH << 16));     // tile_dim1=32, tile_dim2=3
  g1[5] = (int)IMW;                                       // tensor_dim0_stride = 224
  g1[6] = (int)(((unsigned)CH_STRIDE & 0xFFFFu) << 16);   // dim1_stride lo16 @bits[223:208]
  g1[7] = (int)((unsigned)CH_STRIDE >> 16);               // dim1_stride hi32
  g2[0] = CH;                                             // tensor_dim2 = 3
  g2[1] = 0; g2[2] = 0; g2[3] = 0;                        // no dim3, no iterate
  g3[0] = 0; g3[1] = 0; g3[2] = 0; g3[3] = 0;
  gx[0] = 0; gx[1] = 0; gx[2] = 0; gx[3] = 0;             // extra group (unused; zero-filled)
  gx[4] = 0; gx[5] = 0; gx[6] = 0; gx[7] = 0;
  __builtin_amdgcn_tensor_load_to_lds(g0, g1, g2, g3, gx, 0);
}

__device__ __forceinline__ size_t patch_elem_off(unsigned p) {
  unsigned b  = p / NODES;
  unsigned rm = p % NODES;
  unsigned t  = rm / NPF;
  unsigned rr = rm % NPF;
  unsigned gi = rr / GW, gj = rr % GW;
  // video[b, 0, t, gi*32, gj*32]
  return ((size_t)(b * CH) * FRAMES + t) * (size_t)(IMH * IMW)
       + (size_t)(gi * GS) * IMW + (size_t)(gj * GS);
}

// ---------------- main stats kernel ----------------
__global__ void __launch_bounds__(NTHREADS)
stats_kernel(const float* __restrict__ video, float* __restrict__ out, unsigned nblocks) {
  __shared__ float buf[2][SORT_N];    // double-buffered patch + sort arena (32 KB)
  __shared__ float wred[8][4];        // per-wave partial power sums

  const unsigned tid  = threadIdx.x;
  const unsigned wave = tid >> 5;
  const unsigned lane = tid & 31;

  // prologue: prefetch first patch into buf[0] (wave 0 issues; TDM ignores EXEC,
  // other waves skip the whole block via execz branch)
  if (wave == 0) {
    unsigned p0 = blockIdx.x;
    if (p0 < NPATCH)
      tdm_load_patch(video + patch_elem_off(p0), (uint32_t)(uintptr_t)&buf[0][0]);
  }

  unsigned it = 0;
  for (unsigned p = blockIdx.x; p < NPATCH; p += nblocks, ++it) {
    const unsigned par = it & 1u;
    if (wave == 0) {
      unsigned pn = p + nblocks;
      if (pn < NPATCH) {
        // prefetch next patch into the other buffer, then wait for the CURRENT
        // one: TDM ops from one wave complete in order, so tensorcnt<=1 means
        // the older (current) transfer is done.
        tdm_load_patch(video + patch_elem_off(pn), (uint32_t)(uintptr_t)&buf[par ^ 1u][0]);
        __builtin_amdgcn_s_wait_tensorcnt(1);
      } else {
        __builtin_amdgcn_s_wait_tensorcnt(0);
      }
    }
    __syncthreads();                       // buf[par] visible to all waves

    float* s = &buf[par][0];

    // ---- pass 1: power sums S1..S4 (12 elements per thread) ----
    float s1 = 0.f, s2 = 0.f, s3 = 0.f, s4 = 0.f;
#pragma unroll
    for (int q = 0; q < PATCH_N / NTHREADS; ++q) {
      float x  = s[tid + q * NTHREADS];
      float x2 = x * x;
      s1 += x; s2 += x2; s3 += x2 * x; s4 += x2 * x2;
    }
#pragma unroll
    for (int m = 16; m >= 1; m >>= 1) {    // wave32 butterfly
      s1 += __shfl_xor(s1, m, 32);
      s2 += __shfl_xor(s2, m, 32);
      s3 += __shfl_xor(s3, m, 32);
      s4 += __shfl_xor(s4, m, 32);
    }
    if (lane == 0) { wred[wave][0] = s1; wred[wave][1] = s2; wred[wave][2] = s3; wred[wave][3] = s4; }

    // pad sort arena [3072..4096) with +inf
#pragma unroll
    for (int q = 0; q < (SORT_N - PATCH_N) / NTHREADS; ++q)
      s[PATCH_N + tid + q * NTHREADS] = __builtin_inff();

    __syncthreads();

    // ---- reduce 8 wave-partials x 4 moments with two f32 WMMAs ----
    // A = ones(16x4), B rows k = wred[k][*] -> D[0,n] = sum_w wred[w][n]
    float S1 = 0.f, S2 = 0.f, S3 = 0.f, S4 = 0.f;
    if (wave == 0) {
      const unsigned n  = lane & 15u;
      const unsigned nc = n & 3u;          // clamped column: load unconditionally,
      const float    zm = (n < 4u) ? 1.f : 0.f;  // zero out lanes n>=4 via cndmask
      const unsigned hi = lane >> 4;       // A/B vgpr0 carries rows {0,2}, vgpr1 rows {1,3}
      const unsigned r0 = hi ? 2u : 0u;
      const unsigned r1 = hi ? 3u : 1u;
      v2f a;  a[0] = 1.f; a[1] = 1.f;      // A[m,k] = 1 for all m,k
      v2f b03, b47;
      b03[0] = wred[r0][nc]      * zm;
      b03[1] = wred[r1][nc]      * zm;
      b47[0] = wred[r0 + 4u][nc] * zm;
      b47[1] = wred[r1 + 4u][nc] * zm;
      v8f acc = {};
      acc = __builtin_amdgcn_wmma_f32_16x16x4_f32(false, a, false, b03, (short)0, acc, false, false);
      acc = __builtin_amdgcn_wmma_f32_16x16x4_f32(false, a, false, b47, (short)0, acc, false, false);
      // D row m=0 lives in acc[0], lanes 0..15 (N = lane)
      S1 = __shfl(acc[0], 0, 32);
      S2 = __shfl(acc[0], 1, 32);
      S3 = __shfl(acc[0], 2, 32);
      S4 = __shfl(acc[0], 3, 32);
    }

    // ---- moment statistics (kept in thread 0 registers across the sort) ----
    float mean = 0.f, stdv = 0.f, energy = 0.f, rms = 0.f, skew = 0.f, kurt = 0.f;
    if (tid == 0) {
      const float rN = 1.0f / (float)PATCH_N;             // constant reciprocal: no div
      mean = S1 * rN;
      float ex2 = S2 * rN, ex3 = S3 * rN, ex4 = S4 * rN;
      float m2 = ex2 - mean * mean; if (m2 < 0.f) m2 = 0.f;
      stdv   = sqrtf(m2 * ((float)PATCH_N / (float)(PATCH_N - 1)));  // unbiased (torch.std)
      energy = ex2;
      rms    = sqrtf(energy + 1e-8f);
      float m3 = ex3 - 3.f * mean * ex2 + 2.f * mean * mean * mean;
      float m4 = ex4 - 4.f * mean * ex3 + 6.f * mean * mean * ex2
               - 3.f * mean * mean * mean * mean;
      float inv_sd  = 1.0f / (stdv + 1e-8f);              // single runtime divide
      float inv_sd2 = inv_sd * inv_sd;
      skew = m3 * inv_sd2 * inv_sd;
      kurt = m4 * inv_sd2 * inv_sd2 - 3.f;
    }

    // ---- bitonic sort of 4096 elements in LDS ----
    for (unsigned k = 2; k <= SORT_N; k <<= 1) {
      for (unsigned j = k >> 1; j > 0; j >>= 1) {
        __syncthreads();
#pragma unroll 4
        for (unsigned i = tid; i < SORT_N; i += NTHREADS) {
          unsigned ixj = i ^ j;
          if (ixj > i) {
            float va = s[i], vb = s[ixj];
            bool up = ((i & k) == 0u);
            if (up ? (va > vb) : (va < vb)) { s[i] = vb; s[ixj] = va; }
          }
        }
      }
    }
    __syncthreads();

    if (tid == 0) {
      float mn  = s[0];
      float mx  = s[PATCH_N - 1];
      float med = s[(PATCH_N - 1) / 2];                   // lower-middle (torch.median)
      // q25: pos = 0.25*3071 = 767.75 ; q75: pos = 0.75*3071 = 2303.25
      float q25 = s[767]  * 0.25f + s[768]  * 0.75f;
      float q75 = s[2303] * 0.75f + s[2304] * 0.25f;
      float* o = out + (size_t)p * 10;
      o[0] = mean;  o[1] = stdv; o[2] = mn;   o[3] = mx;  o[4] = energy;
      o[5] = rms;   o[6] = skew; o[7] = kurt; o[8] = med; o[9] = q75 - q25;
    }
    // next iteration's TDM overwrites buf[par] only after the sort barrier above,
    // and thread 0's LDS reads precede wave 0's next TDM issue in program order.
  }
}

// ---------------- static edge index (exact reference enumeration order) ----------------
__global__ void edge_kernel(long long* __restrict__ ei) {
  int node = blockIdx.x * blockDim.x + threadIdx.x;
  if (node >= NODES) return;
  int t  = node / NPF;
  int rr = node % NPF;
  int i  = rr / GW, j = rr % GW;

  // f(a): #valid of {a-1,a,a+1} in [0,7)  -> spatial neighbors of (i,j) = f(i)*f(j)-1
  auto fv = [](int a) { return (a == 0 || a == GW - 1) ? 2 : 3; };
  auto gc = [](int tt) {
    int c = 0;
    if (tt + 1 < FRAMES) c++;
    if (tt - 1 >= 0)     c++;
    if (tt + 2 < FRAMES) c++;
    if (tt - 2 >= 0)     c++;
    return c;                           // temporal edge count, same for all nodes of frame tt
  };

  long long off = 0;
  for (int tp = 0; tp < t; ++tp) off += 312 + (long long)NPF * gc(tp); // full earlier frames
  for (int ip = 0; ip < i; ++ip) off += fv(ip) * 19 - GW;              // earlier rows (F=19)
  for (int jp = 0; jp < j; ++jp) off += fv(i) * fv(jp) - 1;            // earlier cols in row
  off += (long long)rr * gc(t);                                        // temporal of earlier nodes

  long long e = off;
  // spatial: di=-1..1, dj=-1..1, skip (0,0)  (reference order)
  for (int di = -1; di <= 1; ++di)
    for (int dj = -1; dj <= 1; ++dj) {
      if (di == 0 && dj == 0) continue;
      int ni = i + di, nj = j + dj;
      if (ni >= 0 && ni < GH && nj >= 0 && nj < GW) {
        ei[e]           = node;
        ei[E_TOTAL + e] = t * NPF + ni * GW + nj;
        ++e;
      }
    }
  // temporal: dt = +1,-1,+2,-2 (reference order)
  const int dts[4] = {1, -1, 2, -2};
  for (int q = 0; q < 4; ++q) {
    int tt = t + dts[q];
    if (tt >= 0 && tt < FRAMES) {
      ei[e]           = node;
      ei[E_TOTAL + e] = tt * NPF + rr;
      ++e;
    }
  }
}

extern "C" void kernel_launch(void* const* d_in, const int* in_sizes, int n_in,
                              void* d_out, int out_size, void* d_ws, size_t ws_size,
                              hipStream_t stream) {
  (void)in_sizes; (void)n_in; (void)out_size; (void)d_ws; (void)ws_size;
  const float* video = (const float*)d_in[0];
  float* out = (float*)d_out;

  const unsigned nblocks = 1024;   // persistent blocks, ~24 patches each, TDM double-buffered
  stats_kernel<<<nblocks, NTHREADS, 0, stream>>>(video, out, nblocks);

  // edge index (int64) concatenated after the 250,880 feature floats (8B aligned)
  long long* ei = (long long*)(out + (size_t)NPATCH * 10);
  edge_kernel<<<(NODES + NTHREADS - 1) / NTHREADS, NTHREADS, 0, stream>>>(ei);
}